// UniGCNconv_18296560681438
// MI455X (gfx1250) — compile-verified
//
#include <hip/hip_runtime.h>
#include <hip/hip_bf16.h>

typedef __attribute__((ext_vector_type(2))) float v2f;
typedef __attribute__((ext_vector_type(8))) float v8f;

#define D 256            // D_IN == D_OUT == 256 (fixed by the reference)
#define LDS_STRIDE 260   // pad to avoid 64-bank conflicts on column reads

// ---------------------------------------------------------------------------
// X = A[M,256] @ W[256,256]  using V_WMMA_F32_16X16X4_F32 (fp32 in, fp32 acc).
// One block = 16 waves. Block handles one 16-row M tile; wave w handles the
// 16-column N tile starting at 16*w. A tile is staged in LDS once per block.
// All addressing is 32-bit (max offset ~102 MB < 2^31).
// ---------------------------------------------------------------------------
__global__ __launch_bounds__(512) void gemm_wmma_f32(
    const float* __restrict__ A, const float* __restrict__ W,
    float* __restrict__ X, int M)
{
    __shared__ float As[16 * LDS_STRIDE];

    const int tid  = threadIdx.x;
    const int wave = tid >> 5;
    const int lane = tid & 31;
    const int row0 = blockIdx.x * 16;

    // Cooperative load of the 16x256 A tile (1024 float4 / 512 threads).
    for (int t = tid; t < 1024; t += 512) {
        const int r  = t >> 6;       // 64 float4 per row
        const int c4 = t & 63;
        int srow = row0 + r;
        if (srow >= M) srow = M - 1;             // clamp (never hit: M%16==0)
        const float4 v = ((const float4*)(A + (unsigned)srow * D))[c4];
        float* dst = &As[r * LDS_STRIDE + c4 * 4];
        dst[0] = v.x; dst[1] = v.y; dst[2] = v.z; dst[3] = v.w;
    }
    __syncthreads();

    // ISA A-frag layout (16x4 f32): lane L -> row m=L&15, K pair = 2*(L>>4).
    const int m     = lane & 15;
    const int kpair = (lane >> 4) * 2;           // 0 or 2
    const int n     = (wave << 4) + (lane & 15); // this wave's output column

    v8f c = {};
    #pragma unroll 8
    for (int k0 = 0; k0 < D; k0 += 4) {
        v2f a, b;
        a.x = As[m * LDS_STRIDE + k0 + kpair];
        a.y = As[m * LDS_STRIDE + k0 + kpair + 1];
        // B frag (4x16 f32): lane L -> col n, rows K = k0+kpair, k0+kpair+1.
        b.x = W[(unsigned)(k0 + kpair) * D + n];
        b.y = W[(unsigned)(k0 + kpair + 1) * D + n];
        c = __builtin_amdgcn_wmma_f32_16x16x4_f32(
                /*neg_a=*/false, a, /*neg_b=*/false, b,
                /*c_mod=*/(short)0, c, /*reuse_a=*/false, /*reuse_b=*/false);
    }

    // C layout: VGPR r -> row m = r + 8*(L>>4), col n = L&15 (of this N tile).
    const int moff = (lane >> 4) * 8;
    const unsigned base = (unsigned)(row0 + moff) * D + (unsigned)n;
    if (row0 + 15 < M) {
        // Fast path: whole tile in range, no per-element EXEC masking.
        #pragma unroll
        for (int r = 0; r < 8; ++r)
            X[base + (unsigned)r * D] = c[r];
    } else {
        #pragma unroll
        for (int r = 0; r < 8; ++r)
            if (row0 + moff + r < M) X[base + (unsigned)r * D] = c[r];
    }
}

// ---------------------------------------------------------------------------
// V -> E: per incidence i, Xe[E[i]] += X[V[i]]  (atomic f32), ecnt[E[i]] += 1.
// 32 threads per incidence row; each lane moves 8 floats (two float4 loads).
// ---------------------------------------------------------------------------
__global__ __launch_bounds__(256) void scatter_v2e(
    const float* __restrict__ X, const int* __restrict__ V,
    const int* __restrict__ E, float* __restrict__ Xe,
    float* __restrict__ ecnt, int nnz)
{
    const long gid = (long)blockIdx.x * 256 + threadIdx.x;
    const long i   = gid >> 5;
    const int lane = (int)(gid & 31);
    if (i >= nnz) return;

    const int v = V[i];
    const int e = E[i];
    const float4* src = (const float4*)(X + (unsigned)v * D) + lane * 2;
    const float4 x0 = src[0];
    const float4 x1 = src[1];
    float* dst = Xe + (unsigned)e * D + lane * 8;
    atomicAdd(dst + 0, x0.x); atomicAdd(dst + 1, x0.y);
    atomicAdd(dst + 2, x0.z); atomicAdd(dst + 3, x0.w);
    atomicAdd(dst + 4, x1.x); atomicAdd(dst + 5, x1.y);
    atomicAdd(dst + 6, x1.z); atomicAdd(dst + 7, x1.w);
    if (lane == 0) atomicAdd(&ecnt[e], 1.0f);
}

// ---------------------------------------------------------------------------
// Xe = (Xe / max(ecnt,1)) * degE   (scatter-mean finalize + degree scaling)
// ---------------------------------------------------------------------------
__global__ __launch_bounds__(256) void edge_finalize(
    float* __restrict__ Xe, const float* __restrict__ ecnt,
    const float* __restrict__ degE, int n_edges)
{
    const long gid = (long)blockIdx.x * 256 + threadIdx.x;
    if (gid >= (long)n_edges * D) return;
    const long e = gid >> 8;                 // D == 256
    const float scale = degE[e] / fmaxf(ecnt[e], 1.0f);
    Xe[gid] *= scale;
}

// ---------------------------------------------------------------------------
// E -> V: per incidence i, out[V[i]] += Xe[E[i]]  (atomic f32).
// Xe is 25.6 MB -> L2-resident gathers.
// ---------------------------------------------------------------------------
__global__ __launch_bounds__(256) void scatter_e2v(
    const float* __restrict__ Xe, const int* __restrict__ V,
    const int* __restrict__ E, float* __restrict__ out, int nnz)
{
    const long gid = (long)blockIdx.x * 256 + threadIdx.x;
    const long i   = gid >> 5;
    const int lane = (int)(gid & 31);
    if (i >= nnz) return;

    const int v = V[i];
    const int e = E[i];
    const float4* src = (const float4*)(Xe + (unsigned)e * D) + lane * 2;
    const float4 x0 = src[0];
    const float4 x1 = src[1];
    float* dst = out + (unsigned)v * D + lane * 8;
    atomicAdd(dst + 0, x0.x); atomicAdd(dst + 1, x0.y);
    atomicAdd(dst + 2, x0.z); atomicAdd(dst + 3, x0.w);
    atomicAdd(dst + 4, x1.x); atomicAdd(dst + 5, x1.y);
    atomicAdd(dst + 6, x1.z); atomicAdd(dst + 7, x1.w);
}

// ---------------------------------------------------------------------------
// out = out * degV + bias
// ---------------------------------------------------------------------------
__global__ __launch_bounds__(256) void node_finalize(
    float* __restrict__ out, const float* __restrict__ degV,
    const float* __restrict__ bias, int n_nodes)
{
    const long gid = (long)blockIdx.x * 256 + threadIdx.x;
    if (gid >= (long)n_nodes * D) return;
    const long n = gid >> 8;
    const int  c = (int)(gid & (D - 1));
    out[gid] = out[gid] * degV[n] + bias[c];
}

extern "C" void kernel_launch(void* const* d_in, const int* in_sizes, int n_in,
                              void* d_out, int out_size, void* d_ws, size_t ws_size,
                              hipStream_t stream) {
    const float* input  = (const float*)d_in[0];   // [N_NODES, 256]
    const int*   V      = (const int*)  d_in[1];   // [NNZ]
    const int*   E      = (const int*)  d_in[2];   // [NNZ]
    const float* degV   = (const float*)d_in[3];   // [N_NODES, 1]
    const float* degE   = (const float*)d_in[4];   // [N_EDGES, 1]
    const float* weight = (const float*)d_in[5];   // [256, 256]
    const float* bias   = (const float*)d_in[6];   // [256]
    float*       out    = (float*)d_out;           // [N_NODES, 256]

    const int nnz     = in_sizes[1];
    const int n_nodes = in_sizes[3];
    const int n_edges = in_sizes[4];

    // Workspace layout: X | Xe | ecnt
    float* X    = (float*)d_ws;
    float* Xe   = X  + (size_t)n_nodes * D;
    float* ecnt = Xe + (size_t)n_edges * D;

    // Zero the atomic accumulators every call (capture-legal async memsets).
    hipMemsetAsync(Xe, 0, ((size_t)n_edges * D + n_edges) * sizeof(float), stream);
    hipMemsetAsync(out, 0, (size_t)n_nodes * D * sizeof(float), stream);

    // 1) X = input @ weight  (fp32 WMMA)
    const int m_tiles = (n_nodes + 15) / 16;
    gemm_wmma_f32<<<m_tiles, 512, 0, stream>>>(input, weight, X, n_nodes);

    // 2) vertex -> edge scatter (numerator + counts)
    {
        const long threads = (long)nnz * 32;
        const int blocks = (int)((threads + 255) / 256);
        scatter_v2e<<<blocks, 256, 0, stream>>>(X, V, E, Xe, ecnt, nnz);
    }

    // 3) edge mean + degE scaling
    {
        const long threads = (long)n_edges * D;
        const int blocks = (int)((threads + 255) / 256);
        edge_finalize<<<blocks, 256, 0, stream>>>(Xe, ecnt, degE, n_edges);
    }

    // 4) edge -> vertex scatter-sum
    {
        const long threads = (long)nnz * 32;
        const int blocks = (int)((threads + 255) / 256);
        scatter_e2v<<<blocks, 256, 0, stream>>>(Xe, V, E, out, nnz);
    }

    // 5) out = out * degV + bias
    {
        const long threads = (long)n_nodes * D;
        const int blocks = (int)((threads + 255) / 256);
        node_finalize<<<blocks, 256, 0, stream>>>(out, degV, bias, n_nodes);
    }
}